// Pooler_12300786336415
// MI455X (gfx1250) — compile-verified
//
#include <hip/hip_runtime.h>
#include <stdint.h>

// FPN ROIAlign pooler for MI455X (gfx1250).
// Strategy: latency-bound gather kernel. One block per (box, 256-thread chunk);
// all threads in a block share one box -> uniform level branch. Output index
// = n*12544 + chunk*256 + tid, fully coalesced NT stores. Feature gathers keep
// adjacent lanes on adjacent bins (same channel) for cacheline reuse; the whole
// 165MB feature set is L2-resident on MI455X (192MB L2), NT output stores avoid
// evicting it.

#define OUTP 7
#define NBIN 49            // 7*7
#define CCH 256
#define PER_BOX (CCH * NBIN)   // 12544 = 49 blocks * 256 threads

__global__ __launch_bounds__(256) void roi_align_fpn_kernel(
    const float* __restrict__ f0, const float* __restrict__ f1,
    const float* __restrict__ f2, const float* __restrict__ f3,
    const float* __restrict__ boxes, const int* __restrict__ bidx,
    float* __restrict__ out, int nboxes)
{
    __shared__ float sbox[4];
    __shared__ int   sb[1];

    const int blk   = blockIdx.x;
    const int n     = blk / NBIN;          // box id (uniform per block)
    const int chunk = blk % NBIN;
    if (n >= nboxes) return;

    // ---- stage box record to LDS via CDNA5 async-to-LDS DMA (ASYNCcnt path)
    if (threadIdx.x == 0) {
        unsigned lds_box = (unsigned)(uintptr_t)&sbox[0];   // low 32 bits = LDS offset
        unsigned lds_bi  = (unsigned)(uintptr_t)&sb[0];
        unsigned long long ga_box = (unsigned long long)(uintptr_t)(boxes + 4ull * n);
        unsigned long long ga_bi  = (unsigned long long)(uintptr_t)(bidx + n);
        asm volatile("global_load_async_to_lds_b128 %0, %1, off"
                     :: "v"(lds_box), "v"(ga_box) : "memory");
        asm volatile("global_load_async_to_lds_b32 %0, %1, off"
                     :: "v"(lds_bi), "v"(ga_bi) : "memory");
    }
#if __has_builtin(__builtin_amdgcn_s_wait_asynccnt)
    __builtin_amdgcn_s_wait_asynccnt(0);
#else
    asm volatile("s_wait_asynccnt 0" ::: "memory");
#endif
    __syncthreads();

    const float x1 = sbox[0], y1 = sbox[1], x2 = sbox[2], y2 = sbox[3];
    const int   b  = sb[0];

    // ---- FPN level mapping (uniform across block)
    const float area = (x2 - x1 + 1.0f) * (y2 - y1 + 1.0f);
    const float s    = sqrtf(area);
    float lvlf = floorf(4.0f + log2f(s * (1.0f / 224.0f) + 1e-6f));
    lvlf = fminf(fmaxf(lvlf, 2.0f), 5.0f);
    const int lvl = (int)lvlf - 2;        // 0..3

    const float* feat; int H, W; float scale;
    if (lvl == 0)      { feat = f0; H = 200; W = 304; scale = 0.25f;    }
    else if (lvl == 1) { feat = f1; H = 100; W = 152; scale = 0.125f;   }
    else if (lvl == 2) { feat = f2; H = 50;  W = 76;  scale = 0.0625f;  }
    else               { feat = f3; H = 25;  W = 38;  scale = 0.03125f; }

    const float rx1  = x1 * scale;
    const float ry1  = y1 * scale;
    const float roiw = fmaxf(x2 * scale - rx1, 1.0f);
    const float roih = fmaxf(y2 * scale - ry1, 1.0f);
    const float binw = roiw * (1.0f / OUTP);
    const float binh = roih * (1.0f / OUTP);

    // ---- this thread's output element: bin fastest for coalescing
    const int idx = chunk * 256 + (int)threadIdx.x;   // < 12544
    const int c   = idx / NBIN;
    const int bin = idx - c * NBIN;
    const int ph  = bin / OUTP;
    const int pw  = bin - ph * OUTP;

    const size_t plane = (size_t)H * (size_t)W;
    const float* __restrict__ fc = feat + (size_t)(b * CCH + c) * plane;

    const float Hf = (float)H, Wf = (float)W;
    const float Hm1 = (float)(H - 1), Wm1 = (float)(W - 1);

    float acc = 0.0f;
    #pragma unroll
    for (int sy = 0; sy < 2; ++sy) {
        const float Y = ry1 + binh * ((float)ph + 0.25f + 0.5f * (float)sy);
        #pragma unroll
        for (int sx = 0; sx < 2; ++sx) {
            const float X = rx1 + binw * ((float)pw + 0.25f + 0.5f * (float)sx);
            const bool valid = (Y >= -1.0f) && (Y <= Hf) && (X >= -1.0f) && (X <= Wf);
            const float yc = fminf(fmaxf(Y, 0.0f), Hm1);
            const float xc = fminf(fmaxf(X, 0.0f), Wm1);
            const int y0 = (int)floorf(yc);
            const int x0 = (int)floorf(xc);
            const int y1i = min(y0 + 1, H - 1);
            const int x1i = min(x0 + 1, W - 1);
            const float ly = yc - (float)y0;
            const float lx = xc - (float)x0;
            const float hy = 1.0f - ly;
            const float hx = 1.0f - lx;
            const float* r0 = fc + (size_t)y0  * (size_t)W;
            const float* r1 = fc + (size_t)y1i * (size_t)W;
            const float v = r0[x0] * (hy * hx) + r0[x1i] * (hy * lx)
                          + r1[x0] * (ly * hx) + r1[x1i] * (ly * lx);
            acc += valid ? v : 0.0f;
        }
    }

    // NT store: output is write-once (51MB); keep L2 for the feature maps.
    __builtin_nontemporal_store(acc * 0.25f, &out[(size_t)n * PER_BOX + (size_t)idx]);
}

extern "C" void kernel_launch(void* const* d_in, const int* in_sizes, int n_in,
                              void* d_out, int out_size, void* d_ws, size_t ws_size,
                              hipStream_t stream) {
    (void)n_in; (void)d_ws; (void)ws_size; (void)out_size;
    const float* f0    = (const float*)d_in[0];
    const float* f1    = (const float*)d_in[1];
    const float* f2    = (const float*)d_in[2];
    const float* f3    = (const float*)d_in[3];
    const float* boxes = (const float*)d_in[4];
    const int*   bidx  = (const int*)d_in[5];
    float* out = (float*)d_out;

    const int nboxes = in_sizes[4] / 4;           // [N,4]
    const int nblocks = nboxes * NBIN;            // one block per (box, 256-elem chunk)
    roi_align_fpn_kernel<<<nblocks, 256, 0, stream>>>(f0, f1, f2, f3, boxes, bidx, out, nboxes);
}